// IPGR_5703716569304
// MI455X (gfx1250) — compile-verified
//
#include <hip/hip_runtime.h>
#include <math.h>

typedef __attribute__((ext_vector_type(2))) float v2f;
typedef __attribute__((ext_vector_type(8))) float v8f;

#define BB 8
#define NN 8192
#define MM 2048
#define MT (MM / 16)   // 128 partial tiles
#define NUM_ITER 4

// ---------------------------------------------------------------------------
// Zero the per-batch max buffer (uint bits of a non-negative float).
// ---------------------------------------------------------------------------
__global__ __launch_bounds__(32) void init_max_kernel(unsigned* maxBuf) {
    if (threadIdx.x < BB) maxBuf[threadIdx.x] = 0u;
}

// ---------------------------------------------------------------------------
// Nearest-neighbor kernel via V_WMMA_F32_16X16X4_F32.
// Block = 256 threads (8 waves), handles one batch slice of 256 query points.
// Grid  = 8 batches * 32 slices = 256 blocks.
//
// A (16x4) rows  = partial points: [-2qx, -2qy, -2qz, |q|^2]   (staged in LDS)
// B (4x16) cols  = query points:   [px, py, pz, 1]
// D[m][n] = |q_m|^2 - 2 q_m . p_n   -> argmin over m == nearest neighbor.
//
// Inner loop is software-pipelined: LDS loads for tiles t+2,t+3 are issued
// while tiles t,t+1 are consumed, hiding ds latency behind ~100 VALU ops.
// ---------------------------------------------------------------------------
__global__ __launch_bounds__(256) void nearest_kernel(
    const float* __restrict__ src,      // refined [8,8192,3]
    const float* __restrict__ partial,  // [8,2048,3]
    int* __restrict__ idxOut,           // [8,8192]
    unsigned* __restrict__ maxBuf)      // [8] float-as-uint max
{
    __shared__ float As[MM * 4];        // 32 KB: {-2x,-2y,-2z,|q|^2} per m

    const int tid   = threadIdx.x;
    const int b     = blockIdx.x >> 5;          // 32 blocks per batch
    const int nbase = (blockIdx.x & 31) * 256;  // 256 queries per block

    // Stage A-table for this batch into LDS.
    const float* pb = partial + b * MM * 3;
    for (int i = tid; i < MM; i += 256) {
        float x = pb[i * 3 + 0];
        float y = pb[i * 3 + 1];
        float z = pb[i * 3 + 2];
        As[i * 4 + 0] = -2.0f * x;
        As[i * 4 + 1] = -2.0f * y;
        As[i * 4 + 2] = -2.0f * z;
        As[i * 4 + 3] = x * x + y * y + z * z;
    }
    __syncthreads();

    const int  lane = tid & 31;
    const int  wave = tid >> 5;
    const bool hiH  = (lane >= 16);
    const int  lc   = lane & 15;

    // Two query tiles per wave (two B operands share every A load).
    const int n0 = nbase + wave * 32 + lc;
    const int n1 = n0 + 16;

    const float* s0 = src + (b * NN + n0) * 3;
    const float* s1 = src + (b * NN + n1) * 3;
    float p0x = s0[0], p0y = s0[1], p0z = s0[2];
    float p1x = s1[0], p1y = s1[1], p1z = s1[2];
    float p2_0 = p0x * p0x + p0y * p0y + p0z * p0z;
    float p2_1 = p1x * p1x + p1y * p1y + p1z * p1z;

    // B layout: lanes 0-15 hold K={0,1}, lanes 16-31 hold K={2,3}.
    v2f bv0, bv1;
    bv0.x = hiH ? p0z : p0x;  bv0.y = hiH ? 1.0f : p0y;
    bv1.x = hiH ? p1z : p1x;  bv1.y = hiH ? 1.0f : p1y;

    v8f zero = {0.f, 0.f, 0.f, 0.f, 0.f, 0.f, 0.f, 0.f};
    v8f rmin0, rmin1;
    int rt0[8], rt1[8];
#pragma unroll
    for (int r = 0; r < 8; ++r) {
        rmin0[r] = 3.0e38f; rmin1[r] = 3.0e38f;
        rt0[r] = 0; rt1[r] = 0;
    }

    // Per-lane base into the LDS A-table (element-major within a tile).
    const float* Abase = As + lc * 4 + (hiH ? 2 : 0);   // + t*64 per tile

    // Software pipeline: keep 2 tiles ahead in flight.
    v2f a0, a1;
    a0.x = Abase[0 * 64 + 0];  a0.y = Abase[0 * 64 + 1];
    a1.x = Abase[1 * 64 + 0];  a1.y = Abase[1 * 64 + 1];

    for (int t = 0; t < MT; t += 2) {
        const int tn = (t + 2) & (MT - 1);   // wraps to 0/1 on final iter (harmless)
        v2f an0, an1;
        an0.x = Abase[tn * 64 + 0];        an0.y = Abase[tn * 64 + 1];
        an1.x = Abase[tn * 64 + 64];       an1.y = Abase[tn * 64 + 65];

        v8f d0 = __builtin_amdgcn_wmma_f32_16x16x4_f32(
            false, a0, false, bv0, (short)0, zero, false, false);
        v8f d1 = __builtin_amdgcn_wmma_f32_16x16x4_f32(
            false, a0, false, bv1, (short)0, zero, false, false);
        v8f e0 = __builtin_amdgcn_wmma_f32_16x16x4_f32(
            false, a1, false, bv0, (short)0, zero, false, false);
        v8f e1 = __builtin_amdgcn_wmma_f32_16x16x4_f32(
            false, a1, false, bv1, (short)0, zero, false, false);

#pragma unroll
        for (int r = 0; r < 8; ++r) {
            if (d0[r] < rmin0[r]) { rmin0[r] = d0[r]; rt0[r] = t; }
            if (d1[r] < rmin1[r]) { rmin1[r] = d1[r]; rt1[r] = t; }
        }
#pragma unroll
        for (int r = 0; r < 8; ++r) {
            if (e0[r] < rmin0[r]) { rmin0[r] = e0[r]; rt0[r] = t + 1; }
            if (e1[r] < rmin1[r]) { rmin1[r] = e1[r]; rt1[r] = t + 1; }
        }

        a0 = an0;
        a1 = an1;
    }

    // In-lane reduction over the 8 D rows. m = t*16 + r (+8 for hi half).
    float best0 = rmin0[0]; int bm0 = rt0[0] * 16;
    float best1 = rmin1[0]; int bm1 = rt1[0] * 16;
#pragma unroll
    for (int r = 1; r < 8; ++r) {
        float v0 = rmin0[r]; int m0 = rt0[r] * 16 + r;
        if (v0 < best0) { best0 = v0; bm0 = m0; }
        float v1 = rmin1[r]; int m1 = rt1[r] * 16 + r;
        if (v1 < best1) { best1 = v1; bm1 = m1; }
    }
    bm0 += hiH ? 8 : 0;
    bm1 += hiH ? 8 : 0;

    // Cross-half combine (lane L <-> lane L^16); prefer smaller m on ties.
    {
        float ov = __shfl_xor(best0, 16, 32);
        int   om = __shfl_xor(bm0, 16, 32);
        if (ov < best0 || (ov == best0 && om < bm0)) { best0 = ov; bm0 = om; }
        ov = __shfl_xor(best1, 16, 32);
        om = __shfl_xor(bm1, 16, 32);
        if (ov < best1 || (ov == best1 && om < bm1)) { best1 = ov; bm1 = om; }
    }

    if (lane < 16) {
        idxOut[b * NN + n0] = bm0;
        idxOut[b * NN + n1] = bm1;
    }

    // Per-batch max of min-distance: wave butterfly + one uint atomic.
    float md0 = sqrtf(fmaxf(best0 + p2_0, 0.0f));
    float md1 = sqrtf(fmaxf(best1 + p2_1, 0.0f));
    float wm  = fmaxf(md0, md1);
#pragma unroll
    for (int s = 16; s > 0; s >>= 1)
        wm = fmaxf(wm, __shfl_xor(wm, s, 32));
    if (lane == 0)
        atomicMax(maxBuf + b, __float_as_uint(wm));
}

// ---------------------------------------------------------------------------
// Lerp update: refined += alpha * (nearest - refined),
// alpha = 0.1 * (2 - d / (max_d + 1e-6)). Recomputes d exactly from the
// chosen neighbor (more accurate than the WMMA-rounded value).
// ---------------------------------------------------------------------------
__global__ __launch_bounds__(256) void update_kernel(
    const float* __restrict__ src,
    const float* __restrict__ partial,
    const int* __restrict__ idxBuf,
    const unsigned* __restrict__ maxBuf,
    float* __restrict__ out)
{
    int i = blockIdx.x * 256 + threadIdx.x;
    if (i >= BB * NN) return;
    int b = i >> 13;  // / 8192

    float rx = src[i * 3 + 0];
    float ry = src[i * 3 + 1];
    float rz = src[i * 3 + 2];

    int m = idxBuf[i];
    const float* q = partial + (b * MM + m) * 3;
    float qx = q[0], qy = q[1], qz = q[2];

    float dx = rx - qx, dy = ry - qy, dz = rz - qz;
    float d  = sqrtf(dx * dx + dy * dy + dz * dz);

    float mx    = __uint_as_float(maxBuf[b]);
    float alpha = 0.1f * (2.0f - d / (mx + 1e-6f));

    out[i * 3 + 0] = rx + alpha * (qx - rx);
    out[i * 3 + 1] = ry + alpha * (qy - ry);
    out[i * 3 + 2] = rz + alpha * (qz - rz);
}

// ---------------------------------------------------------------------------
extern "C" void kernel_launch(void* const* d_in, const int* in_sizes, int n_in,
                              void* d_out, int out_size, void* d_ws, size_t ws_size,
                              hipStream_t stream) {
    (void)in_sizes; (void)n_in; (void)out_size; (void)ws_size;
    const float* pred    = (const float*)d_in[0];   // [8,8192,3]
    const float* partial = (const float*)d_in[1];   // [8,2048,3]
    float*       out     = (float*)d_out;           // [8,8192,3]

    unsigned* maxBuf = (unsigned*)d_ws;                  // 8 words
    int*      idxBuf = (int*)((char*)d_ws + 256);        // 8*8192 ints

    for (int it = 0; it < NUM_ITER; ++it) {
        const float* src = (it == 0) ? pred : out;
        init_max_kernel<<<1, 32, 0, stream>>>(maxBuf);
        nearest_kernel<<<BB * (NN / 256), 256, 0, stream>>>(src, partial, idxBuf, maxBuf);
        update_kernel<<<(BB * NN + 255) / 256, 256, 0, stream>>>(src, partial, idxBuf, maxBuf, out);
    }
}